// RegularGrid_3504693313893
// MI455X (gfx1250) — compile-verified
//
#include <hip/hip_runtime.h>
#include <hip/hip_bf16.h>

// ---------------------------------------------------------------------------
// Problem constants (match the JAX reference)
// ---------------------------------------------------------------------------
#define N_RAYS    4096
#define RESV      128
#define DATA_DIM  28
#define N_INTERS  1536           // RES * 12
#define N_SAMP    1535           // N_INTERS - 1
#define N_CHUNKS  8
#define CHUNK     192            // 8 * 192 = 1536 >= 1535
#define NW        (N_RAYS * N_CHUNKS)   // 32768 compaction chunks
#define N3        (128 * 128 * 128)     // per-channel grid volume
#define HW        (128 * 128)

// VOXEL_LEN = norm([3,3,3]) / 1536 = sqrt(27)/1536 (computed in f32 upstream)
#define VOXEL_LEN (5.19615242270663f / 1536.0f)

typedef __attribute__((ext_vector_type(2))) float v2f;
typedef __attribute__((ext_vector_type(8))) float v8f;

// ---------------------------------------------------------------------------
// Kernel A: per-ray start t + per-chunk valid counts (thread per ray)
// ---------------------------------------------------------------------------
__global__ void ray_setup_kernel(const float* __restrict__ rays_o,
                                 const float* __restrict__ rays_d,
                                 float* __restrict__ starts,
                                 unsigned* __restrict__ cnt)
{
    int r = blockIdx.x * blockDim.x + threadIdx.x;
    if (r >= N_RAYS) return;
    float ox = rays_o[r * 3 + 0], oy = rays_o[r * 3 + 1], oz = rays_o[r * 3 + 2];
    float dx = rays_d[r * 3 + 0], dy = rays_d[r * 3 + 1], dz = rays_d[r * 3 + 2];

    float tx = fminf((1.5f - ox) / dx, (-1.5f - ox) / dx);
    float ty = fminf((1.5f - oy) / dy, (-1.5f - oy) / dy);
    float tz = fminf((1.5f - oz) / dz, (-1.5f - oz) / dz);
    float start = fminf(fmaxf(fmaxf(fmaxf(tx, ty), tz), 0.2f), 6.0f);
    starts[r] = start;

    for (int k = 0; k < N_CHUNKS; ++k) {
        int s0 = k * CHUNK;
        int s1 = min(s0 + CHUNK, N_SAMP);
        unsigned c = 0;
        for (int s = s0; s < s1; ++s) {
            float t  = fmaf((float)s, VOXEL_LEN, start);
            float px = fmaf(t, dx, ox);
            float py = fmaf(t, dy, oy);
            float pz = fmaf(t, dz, oz);
            bool inside = (px > -1.5f) && (px < 1.5f) &&
                          (py > -1.5f) && (py < 1.5f) &&
                          (pz > -1.5f) && (pz < 1.5f);
            c += inside ? 1u : 0u;
        }
        cnt[r * N_CHUNKS + k] = c;
    }
}

// ---------------------------------------------------------------------------
// Kernel B: exclusive scan over 32768 chunk counts (1 block, 1024 thr x 32)
// off[NW] = total number of valid points (NV)
// ---------------------------------------------------------------------------
__global__ void scan_kernel(const unsigned* __restrict__ cnt,
                            unsigned* __restrict__ off)
{
    __shared__ unsigned sh[1024];
    int t = threadIdx.x;
    unsigned local[32];
    unsigned sum = 0;
#pragma unroll
    for (int j = 0; j < 32; ++j) { local[j] = cnt[t * 32 + j]; sum += local[j]; }
    sh[t] = sum;
    __syncthreads();
    for (int d = 1; d < 1024; d <<= 1) {
        unsigned v = (t >= d) ? sh[t - d] : 0u;
        __syncthreads();
        sh[t] += v;
        __syncthreads();
    }
    unsigned base = (t > 0) ? sh[t - 1] : 0u;
#pragma unroll
    for (int j = 0; j < 32; ++j) { off[t * 32 + j] = base; base += local[j]; }
    if (t == 1023) off[NW] = sh[1023];
}

// ---------------------------------------------------------------------------
// Kernel C: main march. One wave per (ray, chunk). Lanes = consecutive
// samples -> coalesced gathers (consecutive samples span ~5 voxels in x).
// Ballot/popc gives the compacted output row for each valid sample.
// ---------------------------------------------------------------------------
__global__ void __launch_bounds__(256)
march_kernel(const float* __restrict__ rays_o,
             const float* __restrict__ rays_d,
             const float* __restrict__ data,
             const float* __restrict__ starts,
             const unsigned* __restrict__ off,
             float* __restrict__ out)
{
    const int tid  = blockIdx.x * blockDim.x + threadIdx.x;
    const int w    = tid >> 5;           // chunk id (32768 waves)
    const int lane = threadIdx.x & 31;
    const int r  = w >> 3;
    const int k  = w & 7;
    const int s0 = k * CHUNK;
    const int s1 = min(s0 + CHUNK, N_SAMP);

    const float ox = rays_o[r * 3 + 0], oy = rays_o[r * 3 + 1], oz = rays_o[r * 3 + 2];
    const float dx = rays_d[r * 3 + 0], dy = rays_d[r * 3 + 1], dz = rays_d[r * 3 + 2];
    const float start = starts[r];

    unsigned row_run  = off[w];
    const unsigned NV = off[NW];
    const size_t mask_base = (size_t)NV * DATA_DIM;
    const size_t sub_base  = mask_base + (size_t)N_RAYS * N_SAMP
                                       + (size_t)N_RAYS * N_INTERS;

    for (int sb = s0; sb < s1; sb += 32) {
        const int s = sb + lane;
        const bool act = s < s1;
        float t  = fmaf((float)s, VOXEL_LEN, start);
        float px = fmaf(t, dx, ox);
        float py = fmaf(t, dy, oy);
        float pz = fmaf(t, dz, oz);
        bool inside = act && (px > -1.5f) && (px < 1.5f) &&
                             (py > -1.5f) && (py < 1.5f) &&
                             (pz > -1.5f) && (pz < 1.5f);
        if (act)
            out[mask_base + (size_t)r * N_SAMP + s] = inside ? 1.0f : 0.0f;

        unsigned bal = (unsigned)__ballot(inside);   // wave32: bits [31:0]
        if (inside) {
            unsigned rank = __popc(bal & ((1u << lane) - 1u));
            size_t row = (size_t)row_run + rank;

            // normalized coords in (-1,1)
            float nx = (px + 1.5f) * (2.0f / 3.0f) - 1.0f;
            float ny = (py + 1.5f) * (2.0f / 3.0f) - 1.0f;
            float nz = (pz + 1.5f) * (2.0f / 3.0f) - 1.0f;

            // sub_pts: grid_pts = (n+1)*64, grid_idx = clip(floor,0,127)
            float gx = (nx + 1.0f) * 64.0f;
            float gy = (ny + 1.0f) * 64.0f;
            float gz = (nz + 1.0f) * 64.0f;
            float gix = fminf(fmaxf(floorf(gx), 0.0f), 127.0f);
            float giy = fminf(fmaxf(floorf(gy), 0.0f), 127.0f);
            float giz = fminf(fmaxf(floorf(gz), 0.0f), 127.0f);
            out[sub_base + row * 3 + 0] = gx - gix;
            out[sub_base + row * 3 + 1] = gy - giy;
            out[sub_base + row * 3 + 2] = gz - giz;

            // trilinear (align_corners=True): coord = (n+1)*0.5*(128-1)
            float xg = (nx + 1.0f) * 63.5f;
            float yg = (ny + 1.0f) * 63.5f;
            float zg = (nz + 1.0f) * 63.5f;
            float xf = floorf(xg), yf = floorf(yg), zf = floorf(zg);
            float fx = xg - xf, fy = yg - yf, fz = zg - zf;
            int ix0 = (int)fminf(fmaxf(xf, 0.0f), 127.0f);
            int iy0 = (int)fminf(fmaxf(yf, 0.0f), 127.0f);
            int iz0 = (int)fminf(fmaxf(zf, 0.0f), 127.0f);
            int ix1 = min(ix0 + 1, 127);
            int iy1 = min(iy0 + 1, 127);
            int iz1 = min(iz0 + 1, 127);

            const int i00 = iz0 * HW + iy0 * 128;
            const int i01 = iz0 * HW + iy1 * 128;
            const int i10 = iz1 * HW + iy0 * 128;
            const int i11 = iz1 * HW + iy1 * 128;

            float* __restrict__ orow = out + row * DATA_DIM;
            const float* g = data;
#pragma unroll 4
            for (int c = 0; c < DATA_DIM; ++c, g += N3) {
                float a00 = g[i00 + ix0], a01 = g[i00 + ix1];
                float a10 = g[i01 + ix0], a11 = g[i01 + ix1];
                float b00 = g[i10 + ix0], b01 = g[i10 + ix1];
                float b10 = g[i11 + ix0], b11 = g[i11 + ix1];
                float x00 = a00 + fx * (a01 - a00);
                float x01 = a10 + fx * (a11 - a10);
                float x10 = b00 + fx * (b01 - b00);
                float x11 = b10 + fx * (b11 - b10);
                float y0  = x00 + fy * (x01 - x00);
                float y1  = x10 + fy * (x11 - x10);
                orow[c] = y0 + fz * (y1 - y0);
            }
        }
        row_run += __popc(bal);
    }
}

// ---------------------------------------------------------------------------
// Kernel W: intersections via v_wmma_f32_16x16x4_f32.
// D[16x16] = A[16x4] x B[4x16]:  A = [start_m, 1, 0, 0] rows,
// B rows = [1...; (tc*16+n)*VOXEL_LEN...; 0; 0]  ->  D[m][n] = start_m + s_n*VL
// ISA f32 16x4 A layout: lanes 0-15 hold K0(v0),K1(v1); lanes 16-31 K2,K3
// (we zero the upper half). C/D: VGPR v -> row v (lanes 0-15) / v+8 (16-31).
// One wave per 16x16 tile; exact launch => EXEC all-ones (WMMA requirement).
// ---------------------------------------------------------------------------
__global__ void __launch_bounds__(256)
intersections_wmma_kernel(const float* __restrict__ starts,
                          const unsigned* __restrict__ off,
                          float* __restrict__ out)
{
    const int tid  = blockIdx.x * blockDim.x + threadIdx.x;
    const int w    = tid >> 5;                 // 24576 tiles
    const int lane = threadIdx.x & 31;
    const int tr   = w / (N_INTERS / 16);      // 256 row tiles (rays)
    const int tc   = w % (N_INTERS / 16);      // 96 col tiles (samples)
    const int nlo  = lane & 15;
    const bool lo  = lane < 16;

    float sv = starts[tr * 16 + nlo];
    v2f a, b;
    a.x = lo ? sv   : 0.0f;                    // K=0 coeff = start_m
    a.y = lo ? 1.0f : 0.0f;                    // K=1 coeff = 1
    b.x = lo ? 1.0f : 0.0f;                    // K=0 row    = 1
    b.y = lo ? (float)(tc * 16 + nlo) * VOXEL_LEN : 0.0f;  // K=1 row = s*VL
    v8f c = {};
    v8f d = __builtin_amdgcn_wmma_f32_16x16x4_f32(
        /*neg_a=*/false, a, /*neg_b=*/false, b,
        /*c_mod=*/(short)0, c, /*reuse_a=*/false, /*reuse_b=*/false);

    const unsigned NV = off[NW];
    const size_t ibase = (size_t)NV * DATA_DIM + (size_t)N_RAYS * N_SAMP;
    const int rowadd = (lane >> 4) * 8;
    float* ob = out + ibase + (size_t)(tr * 16) * N_INTERS + tc * 16 + nlo;
#pragma unroll
    for (int v = 0; v < 8; ++v)
        ob[(size_t)(v + rowadd) * N_INTERS] = d[v];
}

// ---------------------------------------------------------------------------
// Host-side launch (graph-capture safe: only kernel launches on `stream`)
// ---------------------------------------------------------------------------
extern "C" void kernel_launch(void* const* d_in, const int* in_sizes, int n_in,
                              void* d_out, int out_size, void* d_ws, size_t ws_size,
                              hipStream_t stream)
{
    const float* rays_o = (const float*)d_in[0];
    const float* rays_d = (const float*)d_in[1];
    const float* data   = (const float*)d_in[2];
    float* out = (float*)d_out;

    char* ws = (char*)d_ws;
    float*    starts = (float*)ws;                        // 4096 f32   (16 KB)
    unsigned* cnt    = (unsigned*)(ws + 16384);           // 32768 u32  (128 KB)
    unsigned* off    = (unsigned*)(ws + 16384 + 131072);  // 32769 u32

    // A: starts + per-chunk counts
    ray_setup_kernel<<<N_RAYS / 256, 256, 0, stream>>>(rays_o, rays_d, starts, cnt);
    // B: global exclusive scan -> compacted row offsets + NV
    scan_kernel<<<1, 1024, 0, stream>>>(cnt, off);
    // C: mask + compaction + 28-ch trilinear gather + sub_pts (32768 waves)
    march_kernel<<<NW / 8, 256, 0, stream>>>(rays_o, rays_d, data, starts, off, out);
    // W: intersections via WMMA tiles (24576 waves, exact launch)
    intersections_wmma_kernel<<<(N_RAYS / 16) * (N_INTERS / 16) * 32 / 256, 256, 0, stream>>>(
        starts, off, out);
}